// VNAttention_4466765988457
// MI455X (gfx1250) — compile-verified
//
#include <hip/hip_runtime.h>
#include <math.h>

typedef __attribute__((ext_vector_type(2))) float v2f;
typedef __attribute__((ext_vector_type(4))) float v4f;
typedef __attribute__((ext_vector_type(8))) float v8f;

// Problem constants
#define BB 2
#define NN 4096
#define KK 16
#define CC 256
#define HH 8
#define DD 32
#define MM (BB * NN * 3)          // 24576 rows for the linear layers
#define INV_SCALE 0.102062072615966f   // 1/sqrt(32*3)

// ---------------------------------------------------------------------------
// Kernel 1: fused Q/K/V projection.  out[m, o] = sum_c X[m,c] * W[o,c]
// One wave per 16x16 output tile; 16 waves per block share tile_m (16 rows of X).
// Uses V_WMMA_F32_16X16X4_F32, stepping K by 4, 64 iterations, 3 accumulators.
// ---------------------------------------------------------------------------
__global__ __launch_bounds__(512)
void vn_qkv_gemm(const float* __restrict__ X,
                 const float* __restrict__ WQ,
                 const float* __restrict__ WK,
                 const float* __restrict__ WV,
                 float* __restrict__ Q,
                 float* __restrict__ Ko,
                 float* __restrict__ V)
{
    const int wave   = threadIdx.x >> 5;   // 0..15 -> tile_n
    const int lane   = threadIdx.x & 31;
    const int tile_m = blockIdx.x;         // 0..1535
    const int r      = lane & 15;
    const int half   = lane >> 4;
    const int kb     = half * 2;           // K base within 4-wide fragment

    // A fragment source: row (tile_m*16 + r) of X, elements c0+kb, c0+kb+1
    const float* Arow = X + (size_t)(tile_m * 16 + r) * CC + kb;
    // B fragment source: row (tile_n*16 + r) of W (output channel), same K slots
    const int col = wave * 16 + r;
    const float* Bq = WQ + (size_t)col * CC + kb;
    const float* Bk = WK + (size_t)col * CC + kb;
    const float* Bv = WV + (size_t)col * CC + kb;

    v8f accq = {};
    v8f acck = {};
    v8f accv = {};

#pragma unroll 8
    for (int c0 = 0; c0 < CC; c0 += 4) {
        v2f a  = *(const v2f*)(Arow + c0);
        v2f bq = *(const v2f*)(Bq + c0);
        v2f bk = *(const v2f*)(Bk + c0);
        v2f bv = *(const v2f*)(Bv + c0);
        accq = __builtin_amdgcn_wmma_f32_16x16x4_f32(false, a, false, bq, (short)0, accq, false, false);
        acck = __builtin_amdgcn_wmma_f32_16x16x4_f32(false, a, false, bk, (short)0, acck, false, false);
        accv = __builtin_amdgcn_wmma_f32_16x16x4_f32(false, a, false, bv, (short)0, accv, false, false);
    }

    // C/D layout: VGPR rr -> row rr (lanes 0-15) / rr+8 (lanes 16-31), col = lane&15
    const int out_col = wave * 16 + r;
#pragma unroll
    for (int rr = 0; rr < 8; ++rr) {
        size_t off = (size_t)(tile_m * 16 + rr + half * 8) * CC + out_col;
        Q[off]  = accq[rr];
        Ko[off] = acck[rr];
        V[off]  = accv[rr];
    }
}

// ---------------------------------------------------------------------------
// Kernel 2: fused KNN-gather + vector-neuron attention.
// One wave per point (b,n).  Lane owns 8 contiguous channels c = lane*8..+7
// (all inside head h = lane/4).  Logits reduced across the 4-lane head group
// with wave32 shuffles; softmax over K=16 in registers.
// ---------------------------------------------------------------------------
__global__ __launch_bounds__(256)
void vn_attn(const float* __restrict__ Q,
             const float* __restrict__ Kt,
             const float* __restrict__ V,
             const int* __restrict__ knn,
             float* __restrict__ Att)
{
    const int lane  = threadIdx.x & 31;
    const int wid   = blockIdx.x * 8 + (threadIdx.x >> 5);  // 0..8191 = b*N + n
    const int b     = wid >> 12;
    const int cbase = lane * 8;

    // Load this lane's q slice: [3][8]
    float qv[3][8];
    const float* qp = Q + (size_t)wid * 3 * CC + cbase;
#pragma unroll
    for (int s = 0; s < 3; ++s) {
        v4f q0 = *(const v4f*)(qp + s * CC);
        v4f q1 = *(const v4f*)(qp + s * CC + 4);
#pragma unroll
        for (int j = 0; j < 4; ++j) { qv[s][j] = q0[j]; qv[s][j + 4] = q1[j]; }
    }

    // Neighbor indices
    int idx[KK];
    const int* ip = knn + (size_t)wid * KK;
#pragma unroll
    for (int kk = 0; kk < KK; ++kk) idx[kk] = ip[kk];

    // Partial logits over this lane's 24 elements (8 channels x 3 spatial)
    float p[KK];
#pragma unroll
    for (int kk = 0; kk < KK; ++kk) {
        const float* kp = Kt + (size_t)(b * NN + idx[kk]) * 3 * CC + cbase;
        float acc = 0.0f;
#pragma unroll
        for (int s = 0; s < 3; ++s) {
            v4f k0 = *(const v4f*)(kp + s * CC);
            v4f k1 = *(const v4f*)(kp + s * CC + 4);
#pragma unroll
            for (int j = 0; j < 4; ++j) {
                acc = fmaf(qv[s][j],     k0[j], acc);
                acc = fmaf(qv[s][j + 4], k1[j], acc);
            }
        }
        p[kk] = acc;
    }

    // Reduce partial logits within the 4-lane head group (head h = lane/4)
#pragma unroll
    for (int kk = 0; kk < KK; ++kk) {
        p[kk] += __shfl_xor(p[kk], 1, 32);
        p[kk] += __shfl_xor(p[kk], 2, 32);
        p[kk] *= INV_SCALE;
    }

    // Softmax over the 16 neighbors (in registers, identical within head group)
    float mx = p[0];
#pragma unroll
    for (int kk = 1; kk < KK; ++kk) mx = fmaxf(mx, p[kk]);
    float sum = 0.0f;
#pragma unroll
    for (int kk = 0; kk < KK; ++kk) { p[kk] = __expf(p[kk] - mx); sum += p[kk]; }
    const float rsum = 1.0f / sum;

    // Weighted gather-sum of V
    float acc[3][8];
#pragma unroll
    for (int s = 0; s < 3; ++s)
#pragma unroll
        for (int j = 0; j < 8; ++j) acc[s][j] = 0.0f;

#pragma unroll
    for (int kk = 0; kk < KK; ++kk) {
        const float w = p[kk] * rsum;
        const float* vp = V + (size_t)(b * NN + idx[kk]) * 3 * CC + cbase;
#pragma unroll
        for (int s = 0; s < 3; ++s) {
            v4f v0 = *(const v4f*)(vp + s * CC);
            v4f v1 = *(const v4f*)(vp + s * CC + 4);
#pragma unroll
            for (int j = 0; j < 4; ++j) {
                acc[s][j]     = fmaf(w, v0[j], acc[s][j]);
                acc[s][j + 4] = fmaf(w, v1[j], acc[s][j + 4]);
            }
        }
    }

    float* op = Att + (size_t)wid * 3 * CC + cbase;
#pragma unroll
    for (int s = 0; s < 3; ++s) {
        v4f o0, o1;
#pragma unroll
        for (int j = 0; j < 4; ++j) { o0[j] = acc[s][j]; o1[j] = acc[s][j + 4]; }
        *(v4f*)(op + s * CC)     = o0;
        *(v4f*)(op + s * CC + 4) = o1;
    }
}

// ---------------------------------------------------------------------------
// Kernel 3: output projection  out[m,o] = sum_c Att[m,c] * WO[o,c]
// ---------------------------------------------------------------------------
__global__ __launch_bounds__(512)
void vn_out_gemm(const float* __restrict__ A,
                 const float* __restrict__ WO,
                 float* __restrict__ Out)
{
    const int wave   = threadIdx.x >> 5;
    const int lane   = threadIdx.x & 31;
    const int tile_m = blockIdx.x;
    const int r      = lane & 15;
    const int half   = lane >> 4;
    const int kb     = half * 2;

    const float* Arow = A  + (size_t)(tile_m * 16 + r) * CC + kb;
    const float* Brow = WO + (size_t)(wave * 16 + r) * CC + kb;

    v8f acc = {};
#pragma unroll 8
    for (int c0 = 0; c0 < CC; c0 += 4) {
        v2f a = *(const v2f*)(Arow + c0);
        v2f bw = *(const v2f*)(Brow + c0);
        acc = __builtin_amdgcn_wmma_f32_16x16x4_f32(false, a, false, bw, (short)0, acc, false, false);
    }

    const int out_col = wave * 16 + r;
#pragma unroll
    for (int rr = 0; rr < 8; ++rr) {
        size_t off = (size_t)(tile_m * 16 + rr + half * 8) * CC + out_col;
        Out[off] = acc[rr];
    }
}

// ---------------------------------------------------------------------------
extern "C" void kernel_launch(void* const* d_in, const int* in_sizes, int n_in,
                              void* d_out, int out_size, void* d_ws, size_t ws_size,
                              hipStream_t stream) {
    const float* x   = (const float*)d_in[0];   // [B,N,3,C]
    const int*   knn = (const int*)  d_in[1];   // [B,N,K]
    const float* wq  = (const float*)d_in[2];   // [C,C]
    const float* wk  = (const float*)d_in[3];
    const float* wv  = (const float*)d_in[4];
    const float* wo  = (const float*)d_in[5];
    float*       out = (float*)d_out;           // [B,N,3,C]

    const size_t plane = (size_t)MM * CC;       // 6,291,456 floats = 25 MB
    float* q   = (float*)d_ws;
    float* k   = q + plane;
    float* v   = k + plane;
    float* att = v + plane;

    // QKV projections: 1536 tile_m blocks x 16 waves (tile_n)
    vn_qkv_gemm<<<MM / 16, 512, 0, stream>>>(x, wq, wk, wv, q, k, v);
    // Gather + attention: 8192 waves, 8 waves/block
    vn_attn<<<(BB * NN) / 8, 256, 0, stream>>>(q, k, v, knn, att);
    // Output projection
    vn_out_gemm<<<MM / 16, 512, 0, stream>>>(att, wo, out);
}